// SolarisSPModel_14903536517443
// MI455X (gfx1250) — compile-verified
//
#include <hip/hip_runtime.h>
#include <math.h>

#define S_TOK 3520
#define DIMN  1536
#define NHD   12
#define HDN   128
#define FFNN  6144
#define OUTN  64
#define PTOK  880
#define CPAIR 64
#define EPSL  1e-6f

typedef __attribute__((ext_vector_type(2))) float v2f;
typedef __attribute__((ext_vector_type(8))) float v8f;

__device__ __forceinline__ v8f v8zero() {
  v8f z = {0.f,0.f,0.f,0.f,0.f,0.f,0.f,0.f};
  return z;
}

// D = A(16x4) * B(4x16) + C, fp32 WMMA
__device__ __forceinline__ v8f wmma4(v2f a, v2f b, v8f c) {
  return __builtin_amdgcn_wmma_f32_16x16x4_f32(false, a, false, b, (short)0, c, false, false);
}

// ---------------- block reduction (256 threads, wave32) ----------------
__device__ __forceinline__ float blk_reduce(float v, float* sbuf) {
#pragma unroll
  for (int off = 16; off > 0; off >>= 1) v += __shfl_xor(v, off, 32);
  int wid = threadIdx.x >> 5, ln = threadIdx.x & 31;
  if (ln == 0) sbuf[wid] = v;
  __syncthreads();
  float r = 0.f;
  if (threadIdx.x < 8) r = sbuf[threadIdx.x];
  if (threadIdx.x < 32) {
#pragma unroll
    for (int off = 4; off > 0; off >>= 1) r += __shfl_xor(r, off, 32);
  }
  if (threadIdx.x == 0) sbuf[8] = r;
  __syncthreads();
  float out = sbuf[8];
  __syncthreads();
  return out;
}

// ---------------- h = LN(x) * (1 + mod_scale + e_scale) + (mod_shift + e_shift) ----------------
__global__ void k_modulate(const float* __restrict__ x, const float* __restrict__ e,
                           const float* __restrict__ mod, float* __restrict__ out,
                           int ishift, int iscale, int nrows) {
  __shared__ float sbuf[9];
  int s = blockIdx.x;
  int f = s / PTOK;
  const float* xs = x + (size_t)s * DIMN;
  float sum = 0.f, sq = 0.f;
  for (int i = threadIdx.x; i < DIMN; i += 256) { float t = xs[i]; sum += t; sq += t * t; }
  sum = blk_reduce(sum, sbuf);
  sq  = blk_reduce(sq,  sbuf);
  float mean = sum * (1.0f / DIMN);
  float var  = sq  * (1.0f / DIMN) - mean * mean;
  float rstd = rsqrtf(var + EPSL);
  const float* esh = e + ((size_t)f * nrows + ishift) * DIMN;
  const float* esc = e + ((size_t)f * nrows + iscale) * DIMN;
  const float* msh = mod + (size_t)ishift * DIMN;
  const float* msc = mod + (size_t)iscale * DIMN;
  float* os = out + (size_t)s * DIMN;
  for (int i = threadIdx.x; i < DIMN; i += 256) {
    float nrm = (xs[i] - mean) * rstd;
    os[i] = nrm * (1.0f + msc[i] + esc[i]) + (msh[i] + esh[i]);
  }
}

// ---------------- in-place RMSNorm over DIM then rotary (adjacent pairs) ----------------
__global__ void k_rmsrope(float* __restrict__ buf, const float* __restrict__ wnorm,
                          const float* __restrict__ ang) {
  __shared__ float sbuf[9];
  int s = blockIdx.x;
  float* bs = buf + (size_t)s * DIMN;
  float sq = 0.f;
  for (int i = threadIdx.x; i < DIMN; i += 256) { float t = bs[i]; sq += t * t; }
  sq = blk_reduce(sq, sbuf);
  float rstd = rsqrtf(sq * (1.0f / DIMN) + EPSL);
  for (int p = threadIdx.x; p < DIMN / 2; p += 256) {
    int c = p & (CPAIR - 1);
    float a = ang[(size_t)s * CPAIR + c];
    float co = cosf(a), si = sinf(a);
    float t0 = bs[2 * p]     * rstd * wnorm[2 * p];
    float t1 = bs[2 * p + 1] * rstd * wnorm[2 * p + 1];
    bs[2 * p]     = t0 * co - t1 * si;
    bs[2 * p + 1] = t0 * si + t1 * co;
  }
}

// ---------------- C[M,N] = act(A[M,K] @ W[K,N] + bias), fp32 WMMA, double-buffered LDS ----------------
template <int ACT>
__global__ void __launch_bounds__(128) k_gemm(const float* __restrict__ A, const float* __restrict__ W,
                                              const float* __restrict__ bias, float* __restrict__ Cmat,
                                              int M, int N, int K) {
  __shared__ float As[2][64][18];
  __shared__ float Bs[2][64][18];   // transposed: Bs[buf][n][k]
  int m0 = blockIdx.x * 64;
  int n0 = blockIdx.y * 64;
  int tid = threadIdx.x;
  int w = tid >> 5, lane = tid & 31, ln = lane & 15, hi = lane >> 4;

  // per-thread staging indices (fixed across chunks)
  int ia0 = tid,        ia1 = tid + 128;                 // A: 256 float4 of 64x16
  int ra0 = ia0 >> 2,   ca0 = (ia0 & 3) * 4;
  int ra1 = ia1 >> 2,   ca1 = (ia1 & 3) * 4;
  int kb0 = ia0 >> 4,   nb0 = (ia0 & 15) * 4;            // B: 256 float4 of 16x64
  int kb1 = ia1 >> 4,   nb1 = (ia1 & 15) * 4;

  v8f acc[4];
#pragma unroll
  for (int t = 0; t < 4; t++) acc[t] = v8zero();

  // ---- prologue: fetch + stage chunk 0 ----
  float4 av0 = *(const float4*)(A + (size_t)(m0 + ra0) * K + ca0);
  float4 av1 = *(const float4*)(A + (size_t)(m0 + ra1) * K + ca1);
  float4 bv0 = *(const float4*)(W + (size_t)kb0 * N + n0 + nb0);
  float4 bv1 = *(const float4*)(W + (size_t)kb1 * N + n0 + nb1);
  As[0][ra0][ca0 + 0] = av0.x; As[0][ra0][ca0 + 1] = av0.y; As[0][ra0][ca0 + 2] = av0.z; As[0][ra0][ca0 + 3] = av0.w;
  As[0][ra1][ca1 + 0] = av1.x; As[0][ra1][ca1 + 1] = av1.y; As[0][ra1][ca1 + 2] = av1.z; As[0][ra1][ca1 + 3] = av1.w;
  Bs[0][nb0 + 0][kb0] = bv0.x; Bs[0][nb0 + 1][kb0] = bv0.y; Bs[0][nb0 + 2][kb0] = bv0.z; Bs[0][nb0 + 3][kb0] = bv0.w;
  Bs[0][nb1 + 0][kb1] = bv1.x; Bs[0][nb1 + 1][kb1] = bv1.y; Bs[0][nb1 + 2][kb1] = bv1.z; Bs[0][nb1 + 3][kb1] = bv1.w;
  __syncthreads();

  int cur = 0;
  int mrow = 16 * w + ln;
  for (int kc = 16; kc <= K; kc += 16) {
    bool more = kc < K;
    float4 an0, an1, bn0, bn1;
    if (more) {  // prefetch next chunk while computing current
      an0 = *(const float4*)(A + (size_t)(m0 + ra0) * K + kc + ca0);
      an1 = *(const float4*)(A + (size_t)(m0 + ra1) * K + kc + ca1);
      bn0 = *(const float4*)(W + (size_t)(kc + kb0) * N + n0 + nb0);
      bn1 = *(const float4*)(W + (size_t)(kc + kb1) * N + n0 + nb1);
    }
#pragma unroll
    for (int k0 = 0; k0 < 16; k0 += 4) {
      v2f a = *(const v2f*)&As[cur][mrow][k0 + 2 * hi];
#pragma unroll
      for (int t = 0; t < 4; t++) {
        v2f b = *(const v2f*)&Bs[cur][16 * t + ln][k0 + 2 * hi];
        acc[t] = wmma4(a, b, acc[t]);
      }
    }
    if (more) {
      int nxt = cur ^ 1;
      As[nxt][ra0][ca0 + 0] = an0.x; As[nxt][ra0][ca0 + 1] = an0.y; As[nxt][ra0][ca0 + 2] = an0.z; As[nxt][ra0][ca0 + 3] = an0.w;
      As[nxt][ra1][ca1 + 0] = an1.x; As[nxt][ra1][ca1 + 1] = an1.y; As[nxt][ra1][ca1 + 2] = an1.z; As[nxt][ra1][ca1 + 3] = an1.w;
      Bs[nxt][nb0 + 0][kb0] = bn0.x; Bs[nxt][nb0 + 1][kb0] = bn0.y; Bs[nxt][nb0 + 2][kb0] = bn0.z; Bs[nxt][nb0 + 3][kb0] = bn0.w;
      Bs[nxt][nb1 + 0][kb1] = bn1.x; Bs[nxt][nb1 + 1][kb1] = bn1.y; Bs[nxt][nb1 + 2][kb1] = bn1.z; Bs[nxt][nb1 + 3][kb1] = bn1.w;
    }
    __syncthreads();
    cur ^= 1;
  }

#pragma unroll
  for (int t = 0; t < 4; t++) {
#pragma unroll
    for (int r = 0; r < 8; r++) {
      int row = m0 + 16 * w + r + 8 * hi;
      int col = n0 + 16 * t + ln;
      float val = acc[t][r] + bias[col];
      if (ACT == 1) {  // tanh-approx GELU (jax.nn.gelu default)
        float inner = 0.7978845608028654f * (val + 0.044715f * val * val * val);
        val = 0.5f * val * (1.0f + tanhf(inner));
      }
      Cmat[(size_t)row * N + col] = val;
    }
  }
}

// ---------------- block-causal flash attention, one wave per 16-row q tile ----------------
__global__ void __launch_bounds__(128) k_flash(const float* __restrict__ qb, const float* __restrict__ kb,
                                               const float* __restrict__ vb, float* __restrict__ ob) {
  __shared__ float pbuf[4][16][18];
  int head = blockIdx.y;
  int w = threadIdx.x >> 5;
  int lane = threadIdx.x & 31, ln = lane & 15, hi = lane >> 4;
  int qt = blockIdx.x * 4 + w;          // 220 q tiles of 16 rows
  int hoff = head * HDN;

  v2f qf[32];
  {
    const float* qrow = qb + (size_t)(16 * qt + ln) * DIMN + hoff + 2 * hi;
#pragma unroll
    for (int i = 0; i < 32; i++) qf[i] = *(const v2f*)(qrow + 4 * i);
  }
  v8f oacc[8];
#pragma unroll
  for (int t = 0; t < 8; t++) oacc[t] = v8zero();
  float mrow[8], lrow[8];
#pragma unroll
  for (int r = 0; r < 8; r++) { mrow[r] = -3.0e38f; lrow[r] = 0.f; }

  int fq = qt / 55;                      // 880/16 = 55 tiles per frame
  int kmax = 55 * (fq + 1);              // block-causal: frames <= fq fully visible
  const float scale = 0.08838834764831845f;  // 1/sqrt(128)

  for (int j = 0; j < kmax; ++j) {
    v8f sc = v8zero();
    const float* krow = kb + (size_t)(16 * j + ln) * DIMN + hoff + 2 * hi;
#pragma unroll
    for (int i = 0; i < 32; i++) {
      v2f b = *(const v2f*)(krow + 4 * i);
      sc = wmma4(qf[i], b, sc);          // scores 16x16 = Q * K^T
    }
    float corr[8];
#pragma unroll
    for (int r = 0; r < 8; r++) {
      float v = sc[r] * scale;
      float mx = v;
#pragma unroll
      for (int off = 1; off < 16; off <<= 1) mx = fmaxf(mx, __shfl_xor(mx, off, 32));
      float mnew = fmaxf(mrow[r], mx);
      float p = __expf(v - mnew);
      float c = __expf(mrow[r] - mnew);
      float psum = p;
#pragma unroll
      for (int off = 1; off < 16; off <<= 1) psum += __shfl_xor(psum, off, 32);
      lrow[r] = lrow[r] * c + psum;
      mrow[r] = mnew;
      corr[r] = c;
      pbuf[w][r + 8 * hi][ln] = p;       // stage P for A-layout transpose
    }
#pragma unroll
    for (int t = 0; t < 8; t++) {
#pragma unroll
      for (int r = 0; r < 8; r++) oacc[t][r] *= corr[r];
    }
    asm volatile("s_wait_dscnt 0x0" ::: "memory");   // wave-local LDS ordering
    v2f pf[4];
#pragma unroll
    for (int k0 = 0; k0 < 4; k0++) pf[k0] = *(const v2f*)&pbuf[w][ln][4 * k0 + 2 * hi];
    const float* vbase = vb + (size_t)(16 * j) * DIMN + hoff + ln;
#pragma unroll
    for (int t = 0; t < 8; t++) {
#pragma unroll
      for (int k0 = 0; k0 < 4; k0++) {
        int kk = 4 * k0 + 2 * hi;
        v2f b;
        b.x = vbase[(size_t)kk * DIMN + 16 * t];
        b.y = vbase[(size_t)(kk + 1) * DIMN + 16 * t];
        oacc[t] = wmma4(pf[k0], b, oacc[t]);         // O += P * V
      }
    }
  }

  float linv[8];
#pragma unroll
  for (int r = 0; r < 8; r++) linv[r] = 1.0f / lrow[r];
#pragma unroll
  for (int t = 0; t < 8; t++) {
#pragma unroll
    for (int r = 0; r < 8; r++) {
      int row = 16 * qt + r + 8 * hi;
      ob[(size_t)row * DIMN + hoff + 16 * t + ln] = oacc[t][r] * linv[r];
    }
  }
}

// ---------------- x += y * (mod[idx] + e[f,idx]) ----------------
__global__ void k_residual(float* __restrict__ x, const float* __restrict__ y,
                           const float* __restrict__ e, const float* __restrict__ mod, int idx) {
  size_t gid = (size_t)blockIdx.x * 256 + threadIdx.x;
  if (gid >= (size_t)S_TOK * DIMN) return;
  int i = (int)(gid % DIMN);
  int s = (int)(gid / DIMN);
  int f = s / PTOK;
  float g = mod[(size_t)idx * DIMN + i] + e[((size_t)f * 6 + idx) * DIMN + i];
  x[gid] += y[gid] * g;
}

extern "C" void kernel_launch(void* const* d_in, const int* in_sizes, int n_in,
                              void* d_out, int out_size, void* d_ws, size_t ws_size,
                              hipStream_t stream) {
  const float* x    = (const float*)d_in[0];
  const float* e    = (const float*)d_in[1];
  const float* ehd  = (const float*)d_in[2];
  const float* ang  = (const float*)d_in[3];
  const float* modb = (const float*)d_in[4];
  const float* modh = (const float*)d_in[5];
  const float* Wq = (const float*)d_in[6];  const float* bq = (const float*)d_in[7];
  const float* Wk = (const float*)d_in[8];  const float* bk = (const float*)d_in[9];
  const float* Wv = (const float*)d_in[10]; const float* bv = (const float*)d_in[11];
  const float* Wo = (const float*)d_in[12]; const float* bo = (const float*)d_in[13];
  const float* wqn = (const float*)d_in[14]; const float* wkn = (const float*)d_in[15];
  const float* W1 = (const float*)d_in[16]; const float* b1 = (const float*)d_in[17];
  const float* W2 = (const float*)d_in[18]; const float* b2 = (const float*)d_in[19];
  const float* Wh = (const float*)d_in[20]; const float* bh = (const float*)d_in[21];
  float* out = (float*)d_out;
  float* ws  = (float*)d_ws;

  const size_t SD = (size_t)S_TOK * DIMN;
  float* xc  = ws;            // running residual stream
  float* h   = ws + 1 * SD;   // modulated activations / proj output
  float* q   = ws + 2 * SD;
  float* k   = ws + 3 * SD;
  float* v   = ws + 4 * SD;
  float* at  = ws + 5 * SD;   // attention output
  float* mid = ws + 6 * SD;   // FFN hidden [S, FFN]

  hipMemcpyAsync(xc, x, SD * sizeof(float), hipMemcpyDeviceToDevice, stream);

  dim3 b256(256), b128(128);
  dim3 gP(S_TOK / 64, DIMN / 64);

  // ---- attention branch ----
  k_modulate<<<S_TOK, b256, 0, stream>>>(xc, e, modb, h, 0, 1, 6);
  k_gemm<0><<<gP, b128, 0, stream>>>(h, Wq, bq, q, S_TOK, DIMN, DIMN);
  k_gemm<0><<<gP, b128, 0, stream>>>(h, Wk, bk, k, S_TOK, DIMN, DIMN);
  k_gemm<0><<<gP, b128, 0, stream>>>(h, Wv, bv, v, S_TOK, DIMN, DIMN);
  k_rmsrope<<<S_TOK, b256, 0, stream>>>(q, wqn, ang);
  k_rmsrope<<<S_TOK, b256, 0, stream>>>(k, wkn, ang);
  dim3 gA(55, NHD);
  k_flash<<<gA, b128, 0, stream>>>(q, k, v, at);
  k_gemm<0><<<gP, b128, 0, stream>>>(at, Wo, bo, h, S_TOK, DIMN, DIMN);
  int gR = (int)((SD + 255) / 256);
  k_residual<<<gR, b256, 0, stream>>>(xc, h, e, modb, 2);

  // ---- FFN branch ----
  k_modulate<<<S_TOK, b256, 0, stream>>>(xc, e, modb, h, 3, 4, 6);
  dim3 gF1(S_TOK / 64, FFNN / 64);
  k_gemm<1><<<gF1, b128, 0, stream>>>(h, W1, b1, mid, S_TOK, FFNN, DIMN);
  k_gemm<0><<<gP, b128, 0, stream>>>(mid, W2, b2, q, S_TOK, DIMN, FFNN);
  k_residual<<<gR, b256, 0, stream>>>(xc, q, e, modb, 5);

  // ---- output head ----
  k_modulate<<<S_TOK, b256, 0, stream>>>(xc, ehd, modh, h, 0, 1, 2);
  dim3 gH(S_TOK / 64, OUTN / 64);
  k_gemm<0><<<gH, b128, 0, stream>>>(h, Wh, bh, out, S_TOK, OUTN, DIMN);
}